// FeatureEncoding_53979148976774
// MI455X (gfx1250) — compile-verified
//
#include <hip/hip_runtime.h>
#include <hip/hip_bf16.h>

typedef __attribute__((ext_vector_type(2))) float v2f;
typedef __attribute__((ext_vector_type(8))) float v8f;

// Problem constants (from reference setup_inputs)
constexpr int kB = 8;
constexpr int kC = 256;
constexpr int kH = 135;
constexpr int kW = 240;
constexpr int kN = 1024;
constexpr int kP = 5;              // ROI_SIZE
constexpr float kBin = 0.8f;       // BOX / ROI_SIZE
constexpr int kWin = 6;            // all 25 sample corners live in a 6x6 window
constexpr int kKpad = 36;          // 6*6, == 9 * 4 (WMMA K steps)
constexpr int kOutPerCand = kC * kP * kP;  // 6400

// One block per candidate. 256 threads = 8 wave32 waves.
// out[25 pts, 256 ch] = Wmat[25(pad 32), 36] x V[36, 256]  via v_wmma_f32_16x16x4_f32
//
// LDS V layout is K-pair interleaved: element (k, c) lives at
//   s_v[((k>>1)*kC + c)*2 + (k&1)]
// so a lane's WMMA B operand (K = k0, k0+1 with k0 even) is one aligned
// 8-byte contiguous LDS read -> single ds_load_b64, no v_mov assembly.
__global__ __launch_bounds__(256)
void roi_align_wmma_kernel(const float* __restrict__ fm,
                           const float* __restrict__ cand,
                           float* __restrict__ out)
{
    __shared__ float s_w[32 * kKpad];        // Wmat row-major [32][36]     (4608 B)
    __shared__ float s_v[kKpad * kC];        // V patch, K-pair interleaved (36864 B)
                                             // reused afterwards as [c][p] output stage

    const int bid = blockIdx.x;              // b*N + n
    const int tid = threadIdx.x;

    const float cx = cand[(size_t)bid * 2 + 0];
    const float cy = cand[(size_t)bid * 2 + 1];
    const int b = bid / kN;

    // Window base = clamped floor of the first sample point (monotone => min),
    // then pulled back so the full 6x6 window is in-bounds.
    int ry = (int)floorf(fmaxf(cy - 2.0f + 0.5f * kBin, 0.0f));
    int rx = (int)floorf(fmaxf(cx - 2.0f + 0.5f * kBin, 0.0f));
    ry = min(ry, kH - kWin); if (ry < 0) ry = 0;
    rx = min(rx, kW - kWin); if (rx < 0) rx = 0;

    // ---- zero Wmat ----
    for (int i = tid; i < 32 * kKpad; i += 256) s_w[i] = 0.0f;

    // ---- load 6x6xC patch: thread == channel, 36 loads each ----
    {
        const int c = tid;  // 0..255
        const float* fmb = fm + ((size_t)b * kC + c) * (size_t)(kH * kW);
        #pragma unroll
        for (int lr = 0; lr < kWin; ++lr) {
            const float* rowp = fmb + (size_t)(ry + lr) * kW + rx;
            #pragma unroll
            for (int lc = 0; lc < kWin; ++lc) {
                const int k = lr * kWin + lc;
                s_v[((k >> 1) * kC + c) * 2 + (k & 1)] = rowp[lc];
            }
        }
    }
    __syncthreads();   // Wmat zeroed (and patch stored)

    // ---- scatter bilinear weights (torchvision semantics) ----
    if (tid < kP * kP) {
        const int j = tid / kP;          // y index
        const int i = tid % kP;          // x index
        const float x = cx - 2.0f + ((float)i + 0.5f) * kBin;
        const float y = cy - 2.0f + ((float)j + 0.5f) * kBin;
        const bool valid = (y > -1.0f) && (y < (float)kH) && (x > -1.0f) && (x < (float)kW);
        const float yy = fmaxf(y, 0.0f);
        const float xx = fmaxf(x, 0.0f);
        int y0 = (int)floorf(yy);
        int x0 = (int)floorf(xx);
        const bool ycl = (y0 >= kH - 1);
        const bool xcl = (x0 >= kW - 1);
        if (ycl) y0 = kH - 1;
        if (xcl) x0 = kW - 1;
        const int y1 = min(y0 + 1, kH - 1);
        const int x1 = min(x0 + 1, kW - 1);
        const float ly = ycl ? 0.0f : (yy - (float)y0);
        const float lx = xcl ? 0.0f : (xx - (float)x0);
        const float hy = 1.0f - ly, hx = 1.0f - lx;
        const float s = valid ? 1.0f : 0.0f;
        const int r0 = y0 - ry, r1 = y1 - ry;
        const int c0 = x0 - rx, c1 = x1 - rx;
        float* wrow = s_w + tid * kKpad;
        wrow[r0 * kWin + c0] += s * hy * hx;
        wrow[r0 * kWin + c1] += s * hy * lx;
        wrow[r1 * kWin + c0] += s * ly * hx;
        wrow[r1 * kWin + c1] += s * ly * lx;
    }
    __syncthreads();   // Wmat final, V patch resident

    // ---- GEMM: D[32,256] = Wmat[32,36] x V[36,256] via WMMA f32 16x16x4 ----
    const int wave  = tid >> 5;        // 0..7 -> owns channel tiles {2w, 2w+1}
    const int lane  = tid & 31;
    const int lhalf = lane >> 4;       // 0: K pair (0,1)   1: K pair (2,3)
    const int l16   = lane & 15;       // A row / B column within tile

    v8f accs[4];
    #pragma unroll
    for (int mt = 0; mt < 2; ++mt) {
        #pragma unroll
        for (int ntl = 0; ntl < 2; ++ntl) {
            const int nt = wave * 2 + ntl;           // channel tile 0..15
            v8f acc = {};
            #pragma unroll
            for (int kk = 0; kk < kKpad / 4; ++kk) { // 9 K-steps
                const int k0 = kk * 4 + lhalf * 2;   // even
                // A 16x4 layout: lanes 0-15 hold K=0,1; lanes 16-31 hold K=2,3
                const v2f a = *(const v2f*)&s_w[(mt * 16 + l16) * kKpad + k0];
                // B 4x16 layout: one aligned 8B read of the K-pair for this column
                const v2f bb = *(const v2f*)&s_v[((k0 >> 1) * kC + nt * 16 + l16) * 2];
                acc = __builtin_amdgcn_wmma_f32_16x16x4_f32(
                        false, a, false, bb, (short)0, acc, false, false);
            }
            accs[mt * 2 + ntl] = acc;
        }
    }
    __syncthreads();   // everyone done reading s_v -> safe to reuse as output stage

    // ---- stage D into LDS in output order [c][p] ----
    float* s_o = s_v;  // 6400 floats used
    #pragma unroll
    for (int mt = 0; mt < 2; ++mt) {
        #pragma unroll
        for (int ntl = 0; ntl < 2; ++ntl) {
            const int c = (wave * 2 + ntl) * 16 + l16;
            const v8f acc = accs[mt * 2 + ntl];
            #pragma unroll
            for (int v = 0; v < 8; ++v) {
                const int p = mt * 16 + lhalf * 8 + v;   // C/D layout: M = v + 8*half
                if (p < kP * kP) s_o[c * (kP * kP) + p] = acc[v];
            }
        }
    }
    __syncthreads();

    // ---- coalesced 128-bit stores: 6400 floats = 1600 float4 ----
    float4* dst = (float4*)(out + (size_t)bid * kOutPerCand);
    const float4* src = (const float4*)s_o;
    for (int idx = tid; idx < kOutPerCand / 4; idx += 256) {
        dst[idx] = src[idx];
    }
}

extern "C" void kernel_launch(void* const* d_in, const int* in_sizes, int n_in,
                              void* d_out, int out_size, void* d_ws, size_t ws_size,
                              hipStream_t stream) {
    (void)n_in; (void)out_size; (void)d_ws; (void)ws_size;
    const float* fm   = (const float*)d_in[0];
    const float* cand = (const float*)d_in[1];
    float* out        = (float*)d_out;
    const int ncand = in_sizes[1] / 2;   // B*N = 8192
    roi_align_wmma_kernel<<<dim3(ncand), dim3(256), 0, stream>>>(fm, cand, out);
}